// BudgetSampling_987842478307
// MI455X (gfx1250) — compile-verified
//
#include <hip/hip_runtime.h>

// ---------------------------------------------------------------------------
// BudgetSampling for MI455X (gfx1250, wave32)
//
// Strategy: histogram sufficient statistics (1 data pass) -> bisection on the
// histogram (O(bins) once, O(1) per iteration) -> final elementwise pass.
// ~384 MB HBM traffic total instead of ~50+ full passes.
// ---------------------------------------------------------------------------

#define NBINS       8192
#define BINSF       8192.0f
#define M_RATIO     20.0
#define BUDGET_     0.5
#define TOL_        1e-6
#define N_ITER_     1000

typedef __attribute__((ext_vector_type(2))) float v2f;
typedef __attribute__((ext_vector_type(8))) float v8f;

// ---------------------------------------------------------------------------
// Kernel 1: LDS-staged histogram of pq in [0,1): per-bin count and sum.
// 256 blocks x 256 threads, grid-stride over float4s. 64 KB LDS per block.
// ---------------------------------------------------------------------------
__global__ __launch_bounds__(256) void bs_hist(const float* __restrict__ pq, int n,
                                               unsigned int* __restrict__ gcnt,
                                               float* __restrict__ gsum) {
    __shared__ unsigned int lc[NBINS];
    __shared__ float        ls[NBINS];
    for (int b = threadIdx.x; b < NBINS; b += blockDim.x) { lc[b] = 0u; ls[b] = 0.0f; }
    __syncthreads();

    const int tid    = blockIdx.x * blockDim.x + threadIdx.x;
    const int stride = gridDim.x * blockDim.x;
    const int nv     = n >> 2;
    const float4* __restrict__ pq4 = (const float4*)pq;

    for (int i = tid; i < nv; i += stride) {
        // CDNA5 global_prefetch_b8 a few strides ahead (speculative, OOB-safe)
        __builtin_prefetch(&pq4[i + 4 * stride], 0, 0);
        float4 v = pq4[i];
        float e[4] = {v.x, v.y, v.z, v.w};
#pragma unroll
        for (int k = 0; k < 4; ++k) {
            float x = e[k];
            int b = (int)(x * BINSF);
            b = (b < 0) ? 0 : ((b >= NBINS) ? (NBINS - 1) : b);
            atomicAdd(&lc[b], 1u);     // ds_add_u32
            atomicAdd(&ls[b], x);      // ds_add_f32
        }
    }
    // scalar tail (n % 4), handled once
    if (tid == 0) {
        for (int i = nv << 2; i < n; ++i) {
            float x = pq[i];
            int b = (int)(x * BINSF);
            b = (b < 0) ? 0 : ((b >= NBINS) ? (NBINS - 1) : b);
            atomicAdd(&lc[b], 1u);
            atomicAdd(&ls[b], x);
        }
    }
    __syncthreads();
    for (int b = threadIdx.x; b < NBINS; b += blockDim.x) {
        unsigned int c = lc[b];
        if (c) {
            atomicAdd(&gcnt[b], c);
            atomicAdd(&gsum[b], ls[b]);
        }
    }
}

// ---------------------------------------------------------------------------
// Kernel 2: single block. Exclusive scan of the 8192 bins (per-thread chunks
// of 8 + sequential scan of 1024 totals), WMMA grand-total of bin sums
// (v_wmma_f32_16x16x4_f32, B = ones -> row-sum reduction), then the exact
// reference bisection loop evaluated against the histogram.
// ---------------------------------------------------------------------------
__global__ __launch_bounds__(1024) void bs_scan_bisect(const unsigned int* __restrict__ gcnt,
                                                       const float* __restrict__ gsum,
                                                       int n, float* __restrict__ res) {
    __shared__ unsigned int scC[1025];
    __shared__ double       scS[1025];
    __shared__ double       wTot;

    const int t = threadIdx.x;

    // per-thread totals over 8 consecutive bins
    unsigned int cacc = 0; double sacc = 0.0;
    const int base = t << 3;
#pragma unroll
    for (int j = 0; j < 8; ++j) { cacc += gcnt[base + j]; sacc += (double)gsum[base + j]; }
    scC[t] = cacc; scS[t] = sacc;
    __syncthreads();

    if (t == 0) {   // exclusive scan of 1024 chunk totals (trivial cost)
        unsigned int rc = 0; double rs = 0.0;
        for (int i = 0; i < 1024; ++i) {
            unsigned int tc = scC[i]; double ts = scS[i];
            scC[i] = rc; scS[i] = rs;
            rc += tc; rs += ts;
        }
        scC[1024] = rc; scS[1024] = rs;
    }

    // ---- WMMA reduction: grand total of gsum[] via v_wmma_f32_16x16x4_f32 ----
    // A = 16x4 tile of bin sums, B = ones(4x16) => D[m,n] = rowsum(A)[m] + C.
    // Column N=0 of the accumulator (lane 0 + lane 16) carries the full total.
    if (t < 32) {
        const int lane = t;
        const int row  = lane & 15;
        const int kb   = (lane >> 4) << 1;        // lanes 0-15: K=0,1; lanes 16-31: K=2,3
        v8f acc = {};
        v2f bones; bones[0] = 1.0f; bones[1] = 1.0f;
        for (int cb = 0; cb < NBINS; cb += 64) {
            const int idx = cb + row * 4 + kb;
            v2f a; a[0] = gsum[idx]; a[1] = gsum[idx + 1];
            acc = __builtin_amdgcn_wmma_f32_16x16x4_f32(
                false, a, false, bones, (short)0, acc, false, false);
        }
        float s = acc[0] + acc[1] + acc[2] + acc[3] + acc[4] + acc[5] + acc[6] + acc[7];
        float tot = __shfl(s, 0, 32) + __shfl(s, 16, 32);   // column N=0: M=0..7 + M=8..15
        if (lane == 0) wTot = (double)tot;
    }
    __syncthreads();

    // ---- Bisection (mirrors the reference loop exactly) ----
    if (t == 0) {
        const double invN   = 1.0 / (double)n;
        const double totalS = wTot;               // WMMA-computed grand sum(pq)
        float c_min = 1e-7f, c_max = 1e11f;
        float c_med = (c_min + c_max) * 0.5f;
        bool  done  = false;
        for (int it = 0; it < N_ITER_ && !done; ++it) {
            const double c = (double)c_med;
            const double u = M_RATIO / c;          // pq threshold where pqm*c == 1
            double meanv;
            if (u >= 1.0) {
                meanv = (c / M_RATIO) * totalS * invN;   // nothing clips
            } else {
                int b = (int)(u * (double)NBINS);
                if (b >= NBINS) b = NBINS - 1;
                if (b < 0) b = 0;
                const int tb = b >> 3;
                double S = scS[tb];
                unsigned int C = scC[tb];
                for (int j = tb << 3; j < b; ++j) { S += (double)gsum[j]; C += gcnt[j]; }
                const double bs = (double)gsum[b];
                const double bc = (double)gcnt[b];
                const double scale = c / M_RATIO;
                double part = scale * bs;                 // boundary-bin contribution
                if (part > bc) part = bc;
                meanv = (scale * S + part + ((double)n - (double)C - bc)) * invN;
            }
            const double diff = meanv - BUDGET_;
            const bool hi = diff > TOL_;
            const bool lo = diff < -TOL_;
            if (hi) c_max = c_med;
            if (lo) c_min = c_med;
            if (hi || lo) c_med = (c_min + c_max) * 0.5f;
            else done = true;
        }
        const float cfin = (c_med < 1.0f) ? 1.0f : c_med;   // torch.clamp(c, min=1)
        res[0] = cfin;
        res[1] = (float)totalS;
    }
}

// ---------------------------------------------------------------------------
// Kernel 3: out = clip((pq / 20) * c, 0, 1)   (float4 streaming, HBM-bound)
// ---------------------------------------------------------------------------
__global__ __launch_bounds__(256) void bs_out(const float* __restrict__ pq, int n,
                                              const float* __restrict__ res,
                                              float* __restrict__ out) {
    const float c  = res[0];                       // uniform -> scalar load
    const int   i4 = blockIdx.x * blockDim.x + threadIdx.x;
    const int   nv = n >> 2;
    if (i4 < nv) {
        float4 v = ((const float4*)pq)[i4];
        float4 r;
        r.x = fminf(fmaxf((v.x / 20.0f) * c, 0.0f), 1.0f);
        r.y = fminf(fmaxf((v.y / 20.0f) * c, 0.0f), 1.0f);
        r.z = fminf(fmaxf((v.z / 20.0f) * c, 0.0f), 1.0f);
        r.w = fminf(fmaxf((v.w / 20.0f) * c, 0.0f), 1.0f);
        ((float4*)out)[i4] = r;
    }
    if (i4 == 0) {
        for (int i = nv << 2; i < n; ++i)
            out[i] = fminf(fmaxf((pq[i] / 20.0f) * c, 0.0f), 1.0f);
    }
}

// ---------------------------------------------------------------------------
extern "C" void kernel_launch(void* const* d_in, const int* in_sizes, int n_in,
                              void* d_out, int out_size, void* d_ws, size_t ws_size,
                              hipStream_t stream) {
    (void)n_in; (void)out_size; (void)ws_size;
    const float* pq  = (const float*)d_in[0];
    float*       out = (float*)d_out;
    const int    n   = in_sizes[0];

    // workspace layout: [gcnt: NBINS u32][gsum: NBINS f32][res: 2 f32]
    unsigned int* gcnt = (unsigned int*)d_ws;
    float*        gsum = (float*)((char*)d_ws + (size_t)NBINS * 4);
    float*        res  = (float*)((char*)d_ws + (size_t)NBINS * 8);

    // zero the histogram every call (d_ws is not re-poisoned between replays)
    hipMemsetAsync(d_ws, 0, (size_t)NBINS * 8, stream);

    bs_hist<<<256, 256, 0, stream>>>(pq, n, gcnt, gsum);
    bs_scan_bisect<<<1, 1024, 0, stream>>>(gcnt, gsum, n, res);

    const int nv = n >> 2;
    const int blocks3 = (nv + 255) / 256;
    bs_out<<<blocks3 > 0 ? blocks3 : 1, 256, 0, stream>>>(pq, n, res, out);
}